// PearsonLoss_6545530159352
// MI455X (gfx1250) — compile-verified
//
#include <hip/hip_runtime.h>
#include <math.h>

// ---------------------------------------------------------------------------
// Pearson-correlation loss, MI455X (gfx1250, wave32).
// Stage 1: bandwidth-bound row reduction (one row / workgroup, NT b128 loads).
// Stage 2: single-wave WMMA f32 16x16x4 reduction of the 4096 row values.
// ---------------------------------------------------------------------------

typedef float f4  __attribute__((ext_vector_type(4)));
typedef float v2f __attribute__((ext_vector_type(2)));
typedef float v8f __attribute__((ext_vector_type(8)));

#define T_LEN 4096          // row length (reference T)
#define BLOCK 256           // 8 wave32 per workgroup

// --------------------------- Stage 1 ---------------------------------------
// One workgroup per row. Each of 256 lanes streams 4 float4's per input,
// accumulates dot/sx/sy/ssx/ssy, wave32 shfl_xor reduce, LDS combine across
// the 8 waves, thread 0 emits r = num / sqrt(den) for the row.
__global__ __launch_bounds__(BLOCK)
void pearson_rows_kernel(const float* __restrict__ X,
                         const float* __restrict__ Y,
                         float* __restrict__ R) {
    const int row = blockIdx.x;
    const f4* xr = (const f4*)(X + (size_t)row * T_LEN);
    const f4* yr = (const f4*)(Y + (size_t)row * T_LEN);

    float dot = 0.f, sx = 0.f, sy = 0.f, ssx = 0.f, ssy = 0.f;

    #pragma unroll
    for (int i = threadIdx.x; i < T_LEN / 4; i += BLOCK) {
        // read-once data: non-temporal so we don't churn the 192MB L2
        f4 x = __builtin_nontemporal_load(xr + i);
        f4 y = __builtin_nontemporal_load(yr + i);
        dot = fmaf(x.x, y.x, dot); dot = fmaf(x.y, y.y, dot);
        dot = fmaf(x.z, y.z, dot); dot = fmaf(x.w, y.w, dot);
        sx += x.x + x.y + x.z + x.w;
        sy += y.x + y.y + y.z + y.w;
        ssx = fmaf(x.x, x.x, ssx); ssx = fmaf(x.y, x.y, ssx);
        ssx = fmaf(x.z, x.z, ssx); ssx = fmaf(x.w, x.w, ssx);
        ssy = fmaf(y.x, y.x, ssy); ssy = fmaf(y.y, y.y, ssy);
        ssy = fmaf(y.z, y.z, ssy); ssy = fmaf(y.w, y.w, ssy);
    }

    // wave32 reduction (gfx1250: warpSize == 32, start at offset 16)
    #pragma unroll
    for (int off = 16; off >= 1; off >>= 1) {
        dot += __shfl_xor(dot, off);
        sx  += __shfl_xor(sx,  off);
        sy  += __shfl_xor(sy,  off);
        ssx += __shfl_xor(ssx, off);
        ssy += __shfl_xor(ssy, off);
    }

    __shared__ float red[BLOCK / 32][5];
    const int wave = threadIdx.x >> 5;
    const int lane = threadIdx.x & 31;
    if (lane == 0) {
        red[wave][0] = dot; red[wave][1] = sx; red[wave][2] = sy;
        red[wave][3] = ssx; red[wave][4] = ssy;
    }
    __syncthreads();

    if (threadIdx.x == 0) {
        #pragma unroll
        for (int w = 1; w < BLOCK / 32; ++w) {
            dot += red[w][0]; sx += red[w][1]; sy += red[w][2];
            ssx += red[w][3]; ssy += red[w][4];
        }
        const float tf  = (float)T_LEN;
        const float num = tf * dot - sx * sy;
        const float den = (tf * ssx - sx * sx) * (tf * ssy - sy * sy);
        R[row] = num / sqrtf(den);
    }
}

// --------------------------- Stage 2 ---------------------------------------
// Single wave32 block. Sum n row-correlations with V_WMMA_F32_16X16X4_F32:
// D = A(16x4) * ones(4x16) + C  => every column of D accumulates row sums.
// A-layout per ISA 7.12.2: lanes 0-15 hold (M=lane, K=0..1) in VGPR0/1,
// lanes 16-31 hold (M=lane-16, K=2..3). Chunk i covers values [i*64, i*64+64),
// element (m,k) = chunk[m*4 + k].
__global__ __launch_bounds__(32)
void pearson_finalize_kernel(const float* __restrict__ R,
                             float* __restrict__ out, int n) {
    const int lane = threadIdx.x;          // 0..31, EXEC all-1s for WMMA
    const int m  = lane & 15;
    const int kh = (lane >> 4) << 1;       // 0 for lanes 0-15, 2 for 16-31

    v2f ones; ones.x = 1.0f; ones.y = 1.0f;
    v8f c = {};

    const int nIter = n >> 6;              // 64 values per WMMA chunk
    for (int i = 0; i < nIter; ++i) {
        const int base = (i << 6) + (m << 2) + kh;
        v2f a; a.x = R[base]; a.y = R[base + 1];
        // 8 args: (neg_a, A, neg_b, B, c_mod, C, reuse_a, reuse_b)
        c = __builtin_amdgcn_wmma_f32_16x16x4_f32(
                false, a, false, ones, (short)0, c, false, false);
    }

    // cell (m, col) holds the partial sum of stream m (identical per column).
    // In-lane: rows m=0..7 (lanes 0-15) / m=8..15 (lanes 16-31); then fold
    // the two halves of column `lane&15` with one xor-shuffle.
    float s = c[0] + c[1] + c[2] + c[3] + c[4] + c[5] + c[6] + c[7];
    s += __shfl_xor(s, 16);

    // scalar tail if n not a multiple of 64 (not hit for n = 4096)
    if (lane == 0) {
        float tail = 0.f;
        for (int i = nIter << 6; i < n; ++i) tail += R[i];
        out[0] = 1.0f - (s + tail) / (float)n;
    }
}

// ---------------------------------------------------------------------------
extern "C" void kernel_launch(void* const* d_in, const int* in_sizes, int n_in,
                              void* d_out, int out_size, void* d_ws, size_t ws_size,
                              hipStream_t stream) {
    const float* X = (const float*)d_in[0];   // logits [B, 4096] f32
    const float* Y = (const float*)d_in[1];   // target [B, 4096] f32
    float* out = (float*)d_out;               // scalar f32
    float* R   = (float*)d_ws;                // per-row correlations [B]

    const int B = in_sizes[0] / T_LEN;        // 4096 rows

    pearson_rows_kernel<<<dim3(B), dim3(BLOCK), 0, stream>>>(X, Y, R);
    pearson_finalize_kernel<<<dim3(1), dim3(32), 0, stream>>>(R, out, B);
}